// GIN_54674933678411
// MI455X (gfx1250) — compile-verified
//
#include <hip/hip_runtime.h>

#define NNODES 50000
#define MTILES (NNODES / 16)   // 3125, exact

typedef float v2f __attribute__((ext_vector_type(2)));
typedef float v8f __attribute__((ext_vector_type(8)));

// Monotonic bijection float -> uint32 so that unsigned-integer max == float max.
__device__ __forceinline__ unsigned enc_f32(float f) {
    unsigned u = __float_as_uint(f);
    return (u & 0x80000000u) ? ~u : (u | 0x80000000u);
}
// enc value 0 is reserved as "empty" (only -NaN would encode to 0; absent in data).
__device__ __forceinline__ float dec_f32(unsigned e) {
    if (e == 0u) return 0.0f;  // isolated node: DGL/ref fills with 0
    unsigned u = (e & 0x80000000u) ? (e ^ 0x80000000u) : ~e;
    return __uint_as_float(u);
}

__global__ void gin_zero_kernel(unsigned* __restrict__ p, long long n4) {
    long long i = blockIdx.x * (long long)blockDim.x + threadIdx.x;
    if (i < n4) ((uint4*)p)[i] = make_uint4(0u, 0u, 0u, 0u);
}

// One thread per (edge, 4-feature chunk); float4 gather + 4x global_atomic_max_u32.
__global__ void gin_scatter_max_kernel(const int* __restrict__ src, const int* __restrict__ dst,
                                       const float* __restrict__ X, unsigned* __restrict__ agg,
                                       int d, int chunk_shift, long long total) {
    long long idx = blockIdx.x * (long long)blockDim.x + threadIdx.x;
    if (idx >= total) return;
    int e = (int)(idx >> chunk_shift);
    int c = (int)(idx & ((1 << chunk_shift) - 1));
    int s = src[e];
    int t = dst[e];
    const float4 v = ((const float4*)(X + (size_t)s * d))[c];
    unsigned* ap = agg + (size_t)t * d + (size_t)c * 4;
    atomicMax(ap + 0, enc_f32(v.x));
    atomicMax(ap + 1, enc_f32(v.y));
    atomicMax(ap + 2, enc_f32(v.z));
    atomicMax(ap + 3, enc_f32(v.w));
}

// xsum = x + decode(agg)   (float4-vectorized)
__global__ void gin_combine_kernel(const float* __restrict__ X, const unsigned* __restrict__ agg,
                                   float* __restrict__ out, long long n4) {
    long long i = blockIdx.x * (long long)blockDim.x + threadIdx.x;
    if (i >= n4) return;
    float4 x = ((const float4*)X)[i];
    uint4  a = ((const uint4*)agg)[i];
    float4 r;
    r.x = x.x + dec_f32(a.x);
    r.y = x.y + dec_f32(a.y);
    r.z = x.z + dec_f32(a.z);
    r.w = x.w + dec_f32(a.w);
    ((float4*)out)[i] = r;
}

// Y[M,N] = relu?( X[M,K] @ W[K,N] + b[N] ) using V_WMMA_F32_16X16X4_F32.
// Each wave computes a 16(M) x 64(N) strip: one A fragment feeds 4 independent
// accumulators (4 B tiles), cutting row-strided A traffic 4x and giving 4
// independent WMMA chains per K-step. The 4 waves of a block take consecutive
// M-tiles; the tail guard is wave-uniform so EXEC is all-ones inside WMMA.
__global__ __launch_bounds__(128) void gin_gemm_wmma_kernel(
    const float* __restrict__ X, const float* __restrict__ W,
    const float* __restrict__ bias, float* __restrict__ Y,
    int K, int Nout, int relu)
{
    const int lane  = threadIdx.x & 31;
    const int wave  = threadIdx.x >> 5;
    const int mtile = blockIdx.x * 4 + wave;
    if (mtile >= MTILES) return;  // wave-uniform tail guard

    const int row0    = mtile * 16;
    const int l15     = lane & 15;
    const int koff    = (lane >> 4) * 2;          // A/B K split across half-waves
    const int colbase = blockIdx.y * 64 + l15;    // 4 N-tiles: colbase + 16*t

    const float* __restrict__ xrow = X + (size_t)(row0 + l15) * K;

    v8f acc0 = {}, acc1 = {}, acc2 = {}, acc3 = {};
#pragma unroll 2
    for (int k0 = 0; k0 < K; k0 += 4) {
        // A 16x4 f32: lanes 0-15 hold K=k0,k0+1; lanes 16-31 hold K=k0+2,k0+3
        v2f a;
        a.x = xrow[k0 + koff + 0];
        a.y = xrow[k0 + koff + 1];
        // B 4x16 f32 mirrors A across lanes; row-major W[K,N]
        const float* __restrict__ w0 = W + (size_t)(k0 + koff + 0) * Nout + colbase;
        const float* __restrict__ w1 = W + (size_t)(k0 + koff + 1) * Nout + colbase;
        v2f b0, b1, b2, b3;
        b0.x = w0[ 0]; b0.y = w1[ 0];
        b1.x = w0[16]; b1.y = w1[16];
        b2.x = w0[32]; b2.y = w1[32];
        b3.x = w0[48]; b3.y = w1[48];
        acc0 = __builtin_amdgcn_wmma_f32_16x16x4_f32(false, a, false, b0, (short)0, acc0, false, false);
        acc1 = __builtin_amdgcn_wmma_f32_16x16x4_f32(false, a, false, b1, (short)0, acc1, false, false);
        acc2 = __builtin_amdgcn_wmma_f32_16x16x4_f32(false, a, false, b2, (short)0, acc2, false, false);
        acc3 = __builtin_amdgcn_wmma_f32_16x16x4_f32(false, a, false, b3, (short)0, acc3, false, false);
    }

    // C/D 16x16 f32: VGPR r holds M = row0 + r (+8 for upper half-wave), N = col
    const int rbase = row0 + (lane >> 4) * 8;
    v8f accs[4] = {acc0, acc1, acc2, acc3};
#pragma unroll
    for (int t = 0; t < 4; ++t) {
        const int col = colbase + 16 * t;
        const float bv = bias[col];
        float* __restrict__ yc = Y + col;
#pragma unroll
        for (int r = 0; r < 8; ++r) {
            float v = accs[t][r] + bv;
            if (relu) v = fmaxf(v, 0.0f);
            yc[(size_t)(rbase + r) * Nout] = v;
        }
    }
}

static void gin_layer(const int* src, const int* dst, int E,
                      const float* Xin, int d_in_, const float* W, const float* b,
                      int d_out_, float* Y, int relu,
                      unsigned* agg, float* xsum, hipStream_t stream)
{
    long long nelem = (long long)NNODES * d_in_;
    long long n4 = nelem >> 2;
    gin_zero_kernel<<<(unsigned)((n4 + 255) / 256), 256, 0, stream>>>(agg, n4);

    int cs = (d_in_ == 128) ? 5 : 6;  // chunks per edge = d/4
    long long total = (long long)E << cs;
    gin_scatter_max_kernel<<<(unsigned)((total + 255) / 256), 256, 0, stream>>>(
        src, dst, Xin, agg, d_in_, cs, total);

    gin_combine_kernel<<<(unsigned)((n4 + 255) / 256), 256, 0, stream>>>(Xin, agg, xsum, n4);

    dim3 grid((MTILES + 3) / 4, d_out_ / 64);
    gin_gemm_wmma_kernel<<<grid, 128, 0, stream>>>(xsum, W, b, Y, d_in_, d_out_, relu);
}

extern "C" void kernel_launch(void* const* d_in, const int* in_sizes, int n_in,
                              void* d_out, int out_size, void* d_ws, size_t ws_size,
                              hipStream_t stream)
{
    const int*   src = (const int*)d_in[0];
    const int*   dst = (const int*)d_in[1];
    const float* emb = (const float*)d_in[2];
    const float* W1  = (const float*)d_in[3];
    const float* b1  = (const float*)d_in[4];
    const float* W2  = (const float*)d_in[5];
    const float* b2  = (const float*)d_in[6];
    const float* W3  = (const float*)d_in[7];
    const float* b3  = (const float*)d_in[8];
    float* out = (float*)d_out;
    const int E = in_sizes[0];

    const size_t cap = (size_t)NNODES * 256;
    unsigned* agg  = (unsigned*)d_ws;            // encoded segment-max
    float*    xsum = (float*)d_ws + cap;         // x + agg
    float*    h    = (float*)d_ws + 2 * cap;     // layer activations

    // Layer 1: emb (d=128) -> h (d=256), relu
    gin_layer(src, dst, E, emb, 128, W1, b1, 256, h, 1, agg, xsum, stream);
    // Layer 2: h (256) -> h (256), relu (gemm reads xsum only, safe to overwrite h)
    gin_layer(src, dst, E, h, 256, W2, b2, 256, h, 1, agg, xsum, stream);
    // Layer 3: h (256) -> out (128), no relu
    gin_layer(src, dst, E, h, 256, W3, b3, 128, out, 0, agg, xsum, stream);
}